// ASAP_82935818486400
// MI455X (gfx1250) — compile-verified
//
#include <hip/hip_runtime.h>
#include <hip/hip_bf16.h>

#define NPTS   16384      // 128*128 grid points
#define DD     128        // latent dim
#define BATCH  4
#define WPRIME (0.05f/1.05f)
#define CHUNK  16         // nodes prepped per phase-A round (K-chunk = 128 rows)
#define CPB    8          // chunks per block (K per block = 1024 rows)
#define KROWS  (CHUNK*8)  // 128 panel rows per chunk
#define C2     (2*DD + 32) // interleaved pair-row stride (floats); C2%64==32 -> conflict-free
#define TC2    (2*DD)      // global T pair-row stride
#define NS_ITERS 20

// workspace layout (float elements); rowStart ints live at the front
#define OFF_RM 16400                       // after (NPTS+1) ints (padded)
#define OFF_T  (OFF_RM + BATCH*DD*DD)

typedef float v2f __attribute__((ext_vector_type(2)));
typedef float v8f __attribute__((ext_vector_type(8)));

__device__ __forceinline__ v8f v8zero() {
  v8f z = {0.f,0.f,0.f,0.f,0.f,0.f,0.f,0.f};
  return z;
}

// D(16x16,f32) += A(16x4,f32) x B(4x16,f32)
__device__ __forceinline__ v8f wmma4(v2f a, v2f b, v8f c) {
  return __builtin_amdgcn_wmma_f32_16x16x4_f32(false, a, false, b, (short)0, c, false, false);
}

// interleaved panel addressing: element (k, col) -> (k>>1)*stride + 2*col + (k&1)

// ---------------------------------------------------------------- zero init
__global__ void k_zero(float* __restrict__ rm, float* __restrict__ out) {
  int i = blockIdx.x * blockDim.x + threadIdx.x;
  if (i < BATCH * DD * DD) rm[i] = 0.f;
  if (i == 0) out[0] = 0.f;
}

// ------------------------------------------------- CSR row starts (e0 sorted)
__global__ void k_rowstart(const int* __restrict__ e0, int E, int* __restrict__ rowStart) {
  int n = blockIdx.x * blockDim.x + threadIdx.x;
  if (n > NPTS) return;
  int lo = 0, hi = E;
  while (lo < hi) { int mid = (lo + hi) >> 1; if (e0[mid] < n) lo = mid + 1; else hi = mid; }
  rowStart[n] = lo;
}

// ------------------------- fused per-node panel build + split-K WMMA GEMM
__global__ __launch_bounds__(256) void k_asap_gemm(
    const float* __restrict__ x, const float* __restrict__ J,
    const int* __restrict__ e1, const int* __restrict__ rowStart,
    float* __restrict__ Rm)
{
  __shared__ float As[(KROWS/2) * C2];
  __shared__ float Bs[(KROWS/2) * C2];
  const int b    = blockIdx.y;
  const int wave = threadIdx.x >> 5;
  const int lane = threadIdx.x & 31;
  const int half = lane >> 4;
  const int lm   = lane & 15;
  const float* xb = x + (size_t)b * NPTS * 3;
  const float* Jb = J + (size_t)b * NPTS * 3 * DD;

  v8f acc[8];
  #pragma unroll
  for (int j = 0; j < 8; ++j) acc[j] = v8zero();

  for (int cc = 0; cc < CPB; ++cc) {
    const int node0 = (blockIdx.x * CPB + cc) * CHUNK;
    // ---- phase A: each wave builds rows for 2 nodes; lane covers 4 columns
    for (int t = 0; t < 2; ++t) {
      const int nl = wave * 2 + t;
      const int n  = node0 + nl;
      const int s0 = rowStart[n], s1 = rowStart[n + 1];
      const float xn0 = xb[3*n+0], xn1 = xb[3*n+1], xn2 = xb[3*n+2];
      const float* Jn = Jb + (size_t)n * 3 * DD + lane * 4;
      float4 t0 = *(const float4*)(Jn);
      float4 t1 = *(const float4*)(Jn + DD);
      float4 t2 = *(const float4*)(Jn + 2 * DD);
      float jn[3][4] = {{t0.x,t0.y,t0.z,t0.w},{t1.x,t1.y,t1.z,t1.w},{t2.x,t2.y,t2.z,t2.w}};
      float js[3][4] = {};   // sum_m Jb[m]
      float sj[3][4] = {};   // sum_m skew(v_nm)^T Jb[m]
      float hd[4]    = {};   // sum_m v_nm . Jb[m]
      float vs0=0.f, vs1=0.f, vs2=0.f, G=0.f;
      float c00=0.f,c01=0.f,c02=0.f,c11=0.f,c12=0.f,c22=0.f;
      for (int k = s0; k < s1; ++k) {
        const int m = e1[k];
        const float v0 = xn0 - xb[3*m+0], v1 = xn1 - xb[3*m+1], v2 = xn2 - xb[3*m+2];
        const float q = v0*v0 + v1*v1 + v2*v2;
        vs0 += v0; vs1 += v1; vs2 += v2; G += q;
        c00 += q - v0*v0; c11 += q - v1*v1; c22 += q - v2*v2;
        c01 -= v0*v1; c02 -= v0*v2; c12 -= v1*v2;
        const float* Jm = Jb + (size_t)m * 3 * DD + lane * 4;
        float4 u0 = *(const float4*)(Jm);
        float4 u1 = *(const float4*)(Jm + DD);
        float4 u2 = *(const float4*)(Jm + 2 * DD);
        float a0[4]={u0.x,u0.y,u0.z,u0.w}, a1[4]={u1.x,u1.y,u1.z,u1.w}, a2[4]={u2.x,u2.y,u2.z,u2.w};
        #pragma unroll
        for (int c = 0; c < 4; ++c) {
          js[0][c] += a0[c]; js[1][c] += a1[c]; js[2][c] += a2[c];
          sj[0][c] += v2*a1[c] - v1*a2[c];
          sj[1][c] += v0*a2[c] - v2*a0[c];
          sj[2][c] += v1*a0[c] - v0*a1[c];
          hd[c]    += v0*a0[c] + v1*a1[c] + v2*a2[c];
        }
      }
      // symmetric 3x3 inverse of C (matches jnp.linalg.inv, no guard)
      const float m00 = c11*c22 - c12*c12;
      const float m01 = c02*c12 - c01*c22;
      const float m02 = c01*c12 - c02*c11;
      const float det = c00*m00 + c01*m01 + c02*m02;
      const float di  = 1.0f / det;
      const float i00 = m00*di, i01 = m01*di, i02 = m02*di;
      const float i11 = (c00*c22 - c02*c02)*di;
      const float i12 = (c01*c02 - c00*c12)*di;
      const float i22 = (c00*c11 - c01*c01)*di;
      const float gi  = (G < 1e-6f) ? 0.0f : (1.0f / G);
      const float deg = (float)(s1 - s0);
      // emit 8 A-rows / 8 B-rows for this node, interleaved-pair layout,
      // packed as b128 stores covering (2 rows x 2 cols)
      #pragma unroll
      for (int cp = 0; cp < 2; ++cp) {
        float av[2][8], bv[2][8];
        #pragma unroll
        for (int u = 0; u < 2; ++u) {
          const int c = 2*cp + u;
          const float b0 = vs2*jn[1][c] - vs1*jn[2][c] - sj[0][c];
          const float b1 = vs0*jn[2][c] - vs2*jn[0][c] - sj[1][c];
          const float b2 = vs1*jn[0][c] - vs0*jn[1][c] - sj[2][c];
          const float htj = hd[c] - (vs0*jn[0][c] + vs1*jn[1][c] + vs2*jn[2][c]);
          av[u][0]=jn[0][c]; av[u][1]=jn[1][c]; av[u][2]=jn[2][c];
          av[u][3]=b0; av[u][4]=b1; av[u][5]=b2; av[u][6]=htj; av[u][7]=0.f;
          bv[u][0]=2.f*(deg*jn[0][c]-js[0][c]);
          bv[u][1]=2.f*(deg*jn[1][c]-js[1][c]);
          bv[u][2]=2.f*(deg*jn[2][c]-js[2][c]);
          bv[u][3]=-(i00*b0 + i01*b1 + i02*b2);
          bv[u][4]=-(i01*b0 + i11*b1 + i12*b2);
          bv[u][5]=-(i02*b0 + i12*b1 + i22*b2);
          bv[u][6]=-WPRIME*gi*htj;
          bv[u][7]=0.f;
        }
        const int colBase = 2*(lane*4 + 2*cp);
        #pragma unroll
        for (int rp = 0; rp < 4; ++rp) {
          float4 va; va.x=av[0][2*rp]; va.y=av[0][2*rp+1]; va.z=av[1][2*rp]; va.w=av[1][2*rp+1];
          *(float4*)&As[(nl*4 + rp)*C2 + colBase] = va;
          float4 vb; vb.x=bv[0][2*rp]; vb.y=bv[0][2*rp+1]; vb.z=bv[1][2*rp]; vb.w=bv[1][2*rp+1];
          *(float4*)&Bs[(nl*4 + rp)*C2 + colBase] = vb;
        }
      }
    }
    __syncthreads();
    // ---- phase B: Rm += As^T * Bs ; every fragment is one ds_load_b64
    for (int k = 0; k < KROWS; k += 4) {
      const int kp = (k >> 1) + half;
      v2f a = *(const v2f*)&As[kp*C2 + 2*(16*wave + lm)];
      #pragma unroll
      for (int j = 0; j < 8; ++j) {
        v2f bb = *(const v2f*)&Bs[kp*C2 + 2*(16*j + lm)];
        acc[j] = wmma4(a, bb, acc[j]);
      }
    }
    __syncthreads();
  }
  float* rm = Rm + (size_t)b * DD * DD;
  #pragma unroll
  for (int j = 0; j < 8; ++j) {
    #pragma unroll
    for (int r = 0; r < 8; ++r) {
      const int row = 16*wave + r + half * 8;
      const int col = 16*j + lm;
      atomicAdd(&rm[row * DD + col], acc[j][r]);
    }
  }
}

// -------- Newton-Schulz sqrtm trace: sum sqrt(eig(Rm)) = sqrt(s)*tr(sqrtm(Rm/s))
// Y,Z,T are symmetric (polynomials in Rm/s), so one interleaved copy serves
// both A-fragment (via transpose) and B-fragment reads.
__global__ __launch_bounds__(256) void k_ns(
    const float* __restrict__ Rm, float* __restrict__ Tg, float* __restrict__ out)
{
  __shared__ float Ys[(DD/2) * C2];
  __shared__ float Zs[(DD/2) * C2];
  __shared__ float tr[DD];
  const int b = blockIdx.x;
  const float* A = Rm + (size_t)b * DD * DD;
  float* T = Tg + (size_t)b * DD * DD;   // interleaved pair layout, stride TC2
  const int tid = threadIdx.x;
  if (tid < DD) tr[tid] = A[tid * DD + tid];
  __syncthreads();
  if (tid == 0) { float t = 0.f; for (int i = 0; i < DD; ++i) t += tr[i]; tr[0] = t; }
  __syncthreads();
  const float s    = tr[0];
  const float sinv = (s > 0.f) ? 1.0f / s : 0.f;
  for (int idx = tid; idx < DD * DD; idx += 256) {
    const int r = idx >> 7, c = idx & (DD - 1);
    const int a_idx = (r >> 1) * C2 + 2*c + (r & 1);
    Ys[a_idx] = 0.5f * (A[r * DD + c] + A[c * DD + r]) * sinv;  // symmetrize
    Zs[a_idx] = (r == c) ? 1.f : 0.f;
  }
  __syncthreads();
  const int wave = tid >> 5, lane = tid & 31, half = lane >> 4, lm = lane & 15;
  v8f acc[8];
  for (int it = 0; it < NS_ITERS; ++it) {
    // W = Z*Y ; T = 1.5I - 0.5W  (staged via global, L2-resident)
    #pragma unroll
    for (int j = 0; j < 8; ++j) acc[j] = v8zero();
    for (int k = 0; k < DD; k += 4) {
      const int kp = (k >> 1) + half;
      v2f a = *(const v2f*)&Zs[kp*C2 + 2*(16*wave + lm)];  // Z^T == Z
      #pragma unroll
      for (int j = 0; j < 8; ++j) {
        v2f bb = *(const v2f*)&Ys[kp*C2 + 2*(16*j + lm)];
        acc[j] = wmma4(a, bb, acc[j]);
      }
    }
    #pragma unroll
    for (int j = 0; j < 8; ++j) {
      #pragma unroll
      for (int r = 0; r < 8; r += 2) {
        const int row = 16*wave + r + half * 8;
        const int col = 16*j + lm;
        float2 tv;
        tv.x = ((row     == col) ? 1.5f : 0.f) - 0.5f * acc[j][r];
        tv.y = ((row + 1 == col) ? 1.5f : 0.f) - 0.5f * acc[j][r+1];
        *(float2*)&T[(row >> 1) * TC2 + 2*col] = tv;
      }
    }
    __threadfence();
    __syncthreads();
    // Ynew = Y*T
    #pragma unroll
    for (int j = 0; j < 8; ++j) acc[j] = v8zero();
    for (int k = 0; k < DD; k += 4) {
      const int kp = (k >> 1) + half;
      v2f a = *(const v2f*)&Ys[kp*C2 + 2*(16*wave + lm)];  // Y^T == Y
      #pragma unroll
      for (int j = 0; j < 8; ++j) {
        v2f bb = *(const v2f*)&T[kp*TC2 + 2*(16*j + lm)];
        acc[j] = wmma4(a, bb, acc[j]);
      }
    }
    __syncthreads();
    #pragma unroll
    for (int j = 0; j < 8; ++j) {
      #pragma unroll
      for (int r = 0; r < 8; r += 2) {
        const int row = 16*wave + r + half * 8;
        float2 yv; yv.x = acc[j][r]; yv.y = acc[j][r+1];
        *(float2*)&Ys[(row >> 1) * C2 + 2*(16*j + lm)] = yv;
      }
    }
    __syncthreads();
    // Znew = T*Z
    #pragma unroll
    for (int j = 0; j < 8; ++j) acc[j] = v8zero();
    for (int k = 0; k < DD; k += 4) {
      const int kp = (k >> 1) + half;
      v2f a = *(const v2f*)&T[kp*TC2 + 2*(16*wave + lm)];  // T^T == T
      #pragma unroll
      for (int j = 0; j < 8; ++j) {
        v2f bb = *(const v2f*)&Zs[kp*C2 + 2*(16*j + lm)];
        acc[j] = wmma4(a, bb, acc[j]);
      }
    }
    __syncthreads();
    #pragma unroll
    for (int j = 0; j < 8; ++j) {
      #pragma unroll
      for (int r = 0; r < 8; r += 2) {
        const int row = 16*wave + r + half * 8;
        float2 zv; zv.x = acc[j][r]; zv.y = acc[j][r+1];
        *(float2*)&Zs[(row >> 1) * C2 + 2*(16*j + lm)] = zv;
      }
    }
    __syncthreads();
  }
  if (tid < DD) tr[tid] = Ys[(tid >> 1) * C2 + 2*tid + (tid & 1)];
  __syncthreads();
  if (tid == 0) {
    float t = 0.f; for (int i = 0; i < DD; ++i) t += tr[i];
    const float val = (s > 0.f) ? sqrtf(s) * t : 0.f;
    atomicAdd(out, val * (1.0f / BATCH));
  }
}

// ---------------------------------------------------------------- launcher
extern "C" void kernel_launch(void* const* d_in, const int* in_sizes, int n_in,
                              void* d_out, int out_size, void* d_ws, size_t ws_size,
                              hipStream_t stream) {
  (void)n_in; (void)out_size; (void)ws_size;
  const float* x  = (const float*)d_in[0];
  const float* J  = (const float*)d_in[1];
  const int*   e0 = (const int*)d_in[2];
  const int E = in_sizes[2] / 2;
  const int* e1 = e0 + E;

  int*   rowStart = (int*)d_ws;
  float* fws = (float*)d_ws;
  float* Rm  = fws + OFF_RM;
  float* Tg  = fws + OFF_T;
  float* out = (float*)d_out;

  k_zero<<<(BATCH * DD * DD + 255) / 256, 256, 0, stream>>>(Rm, out);
  k_rowstart<<<(NPTS + 1 + 255) / 256, 256, 0, stream>>>(e0, E, rowStart);
  dim3 grid(NPTS / (CHUNK * CPB), BATCH);   // 128 x 4 blocks
  k_asap_gemm<<<grid, 256, 0, stream>>>(x, J, e1, rowStart, Rm);
  k_ns<<<BATCH, 256, 0, stream>>>(Rm, Tg, out);
}